// DihedralModel_61753039782767
// MI455X (gfx1250) — compile-verified
//
#include <hip/hip_runtime.h>
#include <cstdint>
#include <cstddef>

// ---------------------------------------------------------------------------
// Problem constants (from reference)
// ---------------------------------------------------------------------------
#define N_ATOMS       10000
#define N_DIH         8000
#define N_FRAMES      2048
#define D_PER_BLOCK   8
#define THREADS       256
#define FRAMES_PER_TILE (THREADS * 4)        // 1024 frames per block tile
#define BUF_FLOATS    (12 * FRAMES_PER_TILE) // one staging buffer: 48KB
#define SHMEM_BYTES   (2 * BUF_FLOATS * 4)   // double buffer: 96KB (<320KB/WGP)

// ---------------------------------------------------------------------------
// CDNA5 async-copy availability (device pass only; host pass returns 0)
// ---------------------------------------------------------------------------
#if defined(__AMDGCN__) && __has_builtin(__builtin_amdgcn_global_load_async_to_lds_b128)
#define HAVE_ASYNC 1
#else
#define HAVE_ASYNC 0
#endif

#if HAVE_ASYNC
#if __has_builtin(__builtin_amdgcn_s_wait_asynccnt)
#define WAIT_ASYNC(n) __builtin_amdgcn_s_wait_asynccnt(n)
#else
#define WAIT_ASYNC(n) asm volatile("s_wait_asynccnt %0" ::"i"(n) : "memory")
#endif
// Exact pointee type per hipcc diagnostic:
//   param0: '__vector_size__(4*sizeof(int)) int __device__ *'  (AS1)
//   param1: LDS side (AS3)
typedef int v4i_t __attribute__((vector_size(4 * sizeof(int))));
typedef __attribute__((address_space(1))) v4i_t* gptr_b128;
typedef __attribute__((address_space(3))) v4i_t* lptr_b128;
#endif

typedef float v4f __attribute__((ext_vector_type(4)));

__device__ __forceinline__ void normalize3(float& vx, float& vy, float& vz) {
    // matches reference: v / max(sqrt(sum(v*v)), 1e-12)
    float s   = vx * vx + vy * vy + vz * vz;
    float inv = 1.0f / fmaxf(sqrtf(s), 1e-12f);
    vx *= inv; vy *= inv; vz *= inv;
}

// Dihedral math for 4 frames held in P[atom][comp] (v4f), then NT b128 stores.
__device__ __forceinline__ void compute_and_store(const v4f P[4][3],
                                                  float* __restrict__ out,
                                                  int d, int b) {
    v4f o0, o1;
#pragma unroll
    for (int j = 0; j < 4; ++j) {
        float b1x = P[1][0][j] - P[0][0][j];
        float b1y = P[1][1][j] - P[0][1][j];
        float b1z = P[1][2][j] - P[0][2][j];
        float b2x = P[2][0][j] - P[1][0][j];
        float b2y = P[2][1][j] - P[1][1][j];
        float b2z = P[2][2][j] - P[1][2][j];
        float b3x = P[3][0][j] - P[2][0][j];
        float b3y = P[3][1][j] - P[2][1][j];
        float b3z = P[3][2][j] - P[2][2][j];

        normalize3(b1x, b1y, b1z);   // a12
        normalize3(b2x, b2y, b2z);   // a23
        normalize3(b3x, b3y, b3z);   // a34

        float v1x = b1y * b2z - b1z * b2y;   // vp1 = a12 x a23
        float v1y = b1z * b2x - b1x * b2z;
        float v1z = b1x * b2y - b1y * b2x;
        float v2x = b2y * b3z - b2z * b3y;   // vp2 = a23 x a34
        float v2y = b2z * b3x - b2x * b3z;
        float v2z = b2x * b3y - b2y * b3x;
        float v3x = v1y * b2z - v1z * b2y;   // vp3 = vp1 x a23
        float v3y = v1z * b2x - v1x * b2z;
        float v3z = v1x * b2y - v1y * b2x;

        float sp1 = v1x * v2x + v1y * v2y + v1z * v2z;
        float sp2 = v3x * v2x + v3y * v2y + v3z * v2z;

        float inv = 1.0f / fmaxf(sqrtf(sp1 * sp1 + sp2 * sp2), 1e-12f);
        o0[j] = -sp2 * inv;   // out row d
        o1[j] =  sp1 * inv;   // out row N_DIH + d
    }
    // Streamed output (131MB, never re-read): NT stores keep the reused atom
    // rows resident in the 192MB L2.
    __builtin_nontemporal_store(o0, (v4f*)(out + (size_t)d * N_FRAMES + b));
    __builtin_nontemporal_store(o1, (v4f*)(out + (size_t)(N_DIH + d) * N_FRAMES + b));
}

#if HAVE_ASYNC
// Issue 12 async b128 copies (one per row) staging dihedral `ad[0..3]`'s
// 4 atoms x 3 components x 1024-frame tile into `lb`. Each lane copies the
// exact 16B it will later read back -> no cross-wave dependency, no barrier.
__device__ __forceinline__ void issue_rows(const float* __restrict__ x,
                                           const int* __restrict__ ad,
                                           int b, float* lb, int t) {
#pragma unroll
    for (int a = 0; a < 4; ++a) {
        const float* g = x + (size_t)ad[a] * (3 * N_FRAMES) + b;
        float*       l = lb + a * 3 * FRAMES_PER_TILE + t * 4;
#pragma unroll
        for (int c = 0; c < 3; ++c) {
            __builtin_amdgcn_global_load_async_to_lds_b128(
                (gptr_b128)(g + c * N_FRAMES),
                (lptr_b128)(l + c * FRAMES_PER_TILE),
                0, 0);
        }
    }
}
#endif

// Each block: 8 dihedrals x 1024-frame tile; each thread: 4 frames (b128 I/O).
// Input layout x[(atom*3+comp)*N_FRAMES + frame] is frame-contiguous, so
// lane-consecutive frames give perfectly coalesced b128 traffic.
__global__ __launch_bounds__(THREADS)
void dihedral_kernel(const float* __restrict__ x,
                     const int*   __restrict__ atoms,
                     float*       __restrict__ out) {
    const int t  = threadIdx.x;
    const int b  = blockIdx.y * FRAMES_PER_TILE + t * 4;   // frame base (x4)
    const int d0 = blockIdx.x * D_PER_BLOCK;

#if HAVE_ASYNC
    // Double-buffered async-to-LDS pipeline (ASYNCcnt, in-order completion).
    extern __shared__ float lbuf[];
    issue_rows(x, atoms + 4 * d0, b, lbuf, t);             // prime buffer 0

#pragma unroll 1
    for (int i = 0; i < D_PER_BLOCK; ++i) {
        const int d = d0 + i;
        float* cur = lbuf + (i & 1) * BUF_FLOATS;

        if (i + 1 < D_PER_BLOCK) {
            asm volatile("" ::: "memory");
            issue_rows(x, atoms + 4 * (d + 1), b,
                       lbuf + ((i + 1) & 1) * BUF_FLOATS, t);
            WAIT_ASYNC(12);        // <=12 outstanding => batch i fully landed
        } else {
            WAIT_ASYNC(0);
        }
        asm volatile("" ::: "memory");

        v4f P[4][3];
#pragma unroll
        for (int a = 0; a < 4; ++a)
#pragma unroll
            for (int c = 0; c < 3; ++c)
                P[a][c] = *(const v4f*)(cur + (a * 3 + c) * FRAMES_PER_TILE + t * 4);

        compute_and_store(P, out, d, b);
    }
#else
    // Fallback: direct b128 gathers + prefetch of the next dihedral's rows.
#pragma unroll 1
    for (int i = 0; i < D_PER_BLOCK; ++i) {
        const int d = d0 + i;

        if ((i + 1 < D_PER_BLOCK) && ((t & 7) == 0)) {
            const int* an = atoms + 4 * (d + 1);
#pragma unroll
            for (int a = 0; a < 4; ++a) {
                const float* r = x + (size_t)an[a] * (3 * N_FRAMES) + b;
                __builtin_prefetch(r,                0, 3);
                __builtin_prefetch(r + N_FRAMES,     0, 3);
                __builtin_prefetch(r + 2 * N_FRAMES, 0, 3);
            }
        }

        const int* ad = atoms + 4 * d;   // block-uniform -> scalar loads
        v4f P[4][3];
#pragma unroll
        for (int a = 0; a < 4; ++a) {
            const v4f* r = (const v4f*)(x + (size_t)ad[a] * (3 * N_FRAMES) + b);
            P[a][0] = r[0];
            P[a][1] = r[N_FRAMES / 4];
            P[a][2] = r[N_FRAMES / 2];
        }
        compute_and_store(P, out, d, b);
    }
#endif
}

extern "C" void kernel_launch(void* const* d_in, const int* in_sizes, int n_in,
                              void* d_out, int out_size, void* d_ws, size_t ws_size,
                              hipStream_t stream) {
    (void)in_sizes; (void)n_in; (void)out_size; (void)d_ws; (void)ws_size;
    const float* x     = (const float*)d_in[0];   // (N_ATOMS*3*N_FRAMES) f32
    const int*   atoms = (const int*)d_in[1];     // (N_DIH, 4) i32
    float*       out   = (float*)d_out;           // (2*N_DIH, N_FRAMES) f32

    dim3 grid(N_DIH / D_PER_BLOCK, N_FRAMES / FRAMES_PER_TILE);  // (1000, 2)
    dihedral_kernel<<<grid, dim3(THREADS), SHMEM_BYTES, stream>>>(x, atoms, out);
}